// GINModel_8701603742430
// MI455X (gfx1250) — compile-verified
//
#include <hip/hip_runtime.h>
#include <hip/hip_bf16.h>
#include <math.h>

#define DD 64
#define CC 40
#define LL 3
#define BN_EPS 1e-5f

typedef __attribute__((ext_vector_type(2))) float v2f;
typedef __attribute__((ext_vector_type(8))) float v8f;

__device__ __forceinline__ void atomic_add_f32(float* p, float v) {
    __hip_atomic_fetch_add(p, v, __ATOMIC_RELAXED, __HIP_MEMORY_SCOPE_AGENT);
}

// ---------------------------------------------------------------- zero
__global__ void zero_f4_kernel(float4* __restrict__ p, int n4) {
    int i = blockIdx.x * blockDim.x + threadIdx.x;
    if (i < n4) p[i] = make_float4(0.f, 0.f, 0.f, 0.f);
}

// ------------------------------------------------- edge scatter (agg += h[src] into dst)
// one wave32 per edge, 2 floats per lane (D=64)
__global__ __launch_bounds__(256) void gin_scatter_kernel(
    const float* __restrict__ h, const int* __restrict__ ei,
    float* __restrict__ agg, int E)
{
    int gid = blockIdx.x * blockDim.x + threadIdx.x;
    int e = gid >> 5;
    if (e >= E) return;
    int lane = gid & 31;
    int src = ei[e];
    int dst = ei[E + e];
    v2f v = *(const v2f*)(h + (size_t)src * DD + lane * 2);
    float* p = agg + (size_t)dst * DD + lane * 2;
    atomic_add_f32(p,     v.x);
    atomic_add_f32(p + 1, v.y);
}

// ------------------------------------------------- (h + agg) @ W + b, WMMA f32 16x16x4
// block = 128 threads = 4 waves; wave w handles column tile w; blockIdx.x = row tile.
// A-frag: lane L, vgpr v -> M = L%16, K = v + 2*(L/16)
// B-frag: lane L, vgpr v -> K = v + 2*(L/16), N = L%16
// C/D:    lane L, vgpr r -> M = r + 8*(L/16), N = L%16
__global__ __launch_bounds__(128) void gin_matmul_kernel(
    const float* __restrict__ h, const float* __restrict__ agg,
    const float* __restrict__ W, const float* __restrict__ bias,
    float* __restrict__ out, float* __restrict__ colsum,
    float* __restrict__ colsumsq, int n, int doRelu, int doStats)
{
    int rowTile = blockIdx.x;
    int colTile = threadIdx.x >> 5;
    int lane    = threadIdx.x & 31;
    int m  = lane & 15;          // M for A-frag, N for B/C/D-frags
    int kh = lane >> 4;          // selects K pair {2kh, 2kh+1}
    int row = rowTile * 16 + m;
    int rl  = row < n ? row : n - 1;   // clamp loads: keep EXEC all-1 for WMMA
    int col = colTile * 16 + m;

    const float* hp = h   + (size_t)rl * DD;
    const float* ap = agg + (size_t)rl * DD;

    v8f c = {};
#pragma unroll
    for (int k0 = 0; k0 < DD; k0 += 4) {
        int ka = k0 + 2 * kh;
        v2f hv = *(const v2f*)(hp + ka);
        v2f av = *(const v2f*)(ap + ka);
        v2f a; a.x = hv.x + av.x; a.y = hv.y + av.y;
        v2f b; b.x = W[(size_t)ka * DD + col];
               b.y = W[(size_t)(ka + 1) * DD + col];
        c = __builtin_amdgcn_wmma_f32_16x16x4_f32(
                false, a, false, b, (short)0, c, false, false);
    }

    float bcol = bias[col];
    float s = 0.f, s2 = 0.f;
#pragma unroll
    for (int r = 0; r < 8; ++r) {
        int outRow = rowTile * 16 + r + 8 * kh;
        float val = c[r] + bcol;
        if (doRelu) val = fmaxf(val, 0.f);
        if (outRow < n) {
            out[(size_t)outRow * DD + col] = val;
            s  += val;
            s2 += val * val;
        }
    }
    if (doStats) {
        s  += __shfl_xor(s, 16, 32);   // fold the two half-wave copies of this column
        s2 += __shfl_xor(s2, 16, 32);
        if (kh == 0) {
            atomic_add_f32(colsum   + col, s);
            atomic_add_f32(colsumsq + col, s2);
        }
    }
}

// ------------------------------------------------- BN finalize: scale/shift per column
__global__ void bn_finalize_kernel(
    const float* __restrict__ colsum, const float* __restrict__ colsumsq,
    const float* __restrict__ gamma, const float* __restrict__ beta,
    float* __restrict__ scale, float* __restrict__ shift, int n)
{
    int c = threadIdx.x;
    if (c >= DD) return;
    float invN = 1.f / (float)n;
    float mu   = colsum[c] * invN;
    float var  = colsumsq[c] * invN - mu * mu;
    float sc   = gamma[c] * rsqrtf(var + BN_EPS);
    scale[c] = sc;
    shift[c] = beta[c] - mu * sc;
}

// ------------------------------------------------- BN apply (elementwise, float4)
__global__ __launch_bounds__(256) void bn_apply_kernel(
    const float* __restrict__ hin, const float* __restrict__ scale,
    const float* __restrict__ shift, float* __restrict__ hout, int total4)
{
    int i = blockIdx.x * blockDim.x + threadIdx.x;
    if (i >= total4) return;
    int c = (i & 15) * 4;                    // 64 cols / 4 = 16 vec4 per row
    float4 v = ((const float4*)hin)[i];
    v.x = v.x * scale[c + 0] + shift[c + 0];
    v.y = v.y * scale[c + 1] + shift[c + 1];
    v.z = v.z * scale[c + 2] + shift[c + 2];
    v.w = v.w * scale[c + 3] + shift[c + 3];
    ((float4*)hout)[i] = v;
}

// ------------------------------------------------- pad W_final (64x40 -> 64x64), b_final
__global__ void pad_final_kernel(const float* __restrict__ Wf,
                                 const float* __restrict__ bf,
                                 float* __restrict__ Wpad,
                                 float* __restrict__ bpad)
{
    int i = blockIdx.x * blockDim.x + threadIdx.x;
    if (i < DD * DD) {
        int r = i >> 6, c = i & 63;
        Wpad[i] = (c < CC) ? Wf[r * CC + c] : 0.f;
    }
    if (i < DD) bpad[i] = (i < CC) ? bf[i] : 0.f;
}

// ------------------------------------------------- log_softmax over 40 classes, wave/row
__global__ __launch_bounds__(256) void logsoftmax_kernel(
    const float* __restrict__ y, float* __restrict__ out, int n)
{
    int gid  = blockIdx.x * blockDim.x + threadIdx.x;
    int row  = gid >> 5;
    int lane = gid & 31;
    if (row >= n) return;
    const float* yr = y + (size_t)row * DD;
    float v0 = yr[lane];                               // lanes 0..31 all < 40
    bool  h1 = (lane + 32) < CC;                       // lanes 0..7
    float v1 = h1 ? yr[lane + 32] : -3.402823466e38f;
    float mx = fmaxf(v0, v1);
    for (int o = 16; o > 0; o >>= 1) mx = fmaxf(mx, __shfl_xor(mx, o, 32));
    float s = expf(v0 - mx) + (h1 ? expf(v1 - mx) : 0.f);
    for (int o = 16; o > 0; o >>= 1) s += __shfl_xor(s, o, 32);
    float lse = mx + logf(s);
    out[(size_t)row * CC + lane] = v0 - lse;
    if (h1) out[(size_t)row * CC + lane + 32] = v1 - lse;
}

// ----------------------------------------------------------------------------
extern "C" void kernel_launch(void* const* d_in, const int* in_sizes, int n_in,
                              void* d_out, int out_size, void* d_ws, size_t ws_size,
                              hipStream_t stream) {
    const float* x       = (const float*)d_in[0];
    const int*   ei      = (const int*)  d_in[1];
    const float* W_proj  = (const float*)d_in[2];
    const float* b_proj  = (const float*)d_in[3];
    const float* Ws      = (const float*)d_in[4];
    const float* bs      = (const float*)d_in[5];
    const float* W_final = (const float*)d_in[6];
    const float* b_final = (const float*)d_in[7];
    const float* gammas  = (const float*)d_in[8];
    const float* betas   = (const float*)d_in[9];
    float* out = (float*)d_out;

    int n = in_sizes[0] / DD;     // 100000
    int E = in_sizes[1] / 2;      // 1600000

    size_t nd = (size_t)n * DD;
    float* ws    = (float*)d_ws;
    float* h_cur = ws;
    float* h_tmp = ws + nd;
    float* agg   = ws + 2 * nd;
    float* stats = ws + 3 * nd;          // colsum[64] + colsumsq[64]
    float* scale = stats + 128;
    float* shift = scale + DD;
    float* Wpad  = shift + DD;           // 4096
    float* bpad  = Wpad + DD * DD;       // 64

    int nRowTiles     = (n + 15) / 16;
    int nd4           = (int)(nd / 4);
    int zeroBlocks    = (nd4 + 255) / 256;
    int scatterBlocks = (int)(((long long)E * 32 + 255) / 256);
    int lsBlocks      = (int)(((long long)n * 32 + 255) / 256);

    for (int layer = 0; layer <= LL; ++layer) {
        const float* hin = (layer == 0) ? x : h_cur;
        const float* W   = (layer == 0) ? W_proj : Ws + (size_t)(layer - 1) * DD * DD;
        const float* b   = (layer == 0) ? b_proj : bs + (size_t)(layer - 1) * DD;

        zero_f4_kernel<<<zeroBlocks, 256, 0, stream>>>((float4*)agg, nd4);
        zero_f4_kernel<<<1, 32, 0, stream>>>((float4*)stats, 32);
        gin_scatter_kernel<<<scatterBlocks, 256, 0, stream>>>(hin, ei, agg, E);
        gin_matmul_kernel<<<nRowTiles, 128, 0, stream>>>(
            hin, agg, W, b, h_tmp, stats, stats + DD, n, /*relu*/1, /*stats*/1);
        bn_finalize_kernel<<<1, DD, 0, stream>>>(
            stats, stats + DD, gammas + (size_t)layer * DD, betas + (size_t)layer * DD,
            scale, shift, n);
        bn_apply_kernel<<<zeroBlocks, 256, 0, stream>>>(h_tmp, scale, shift, h_cur, nd4);
    }

    // final conv (D -> C, padded to 64) + log_softmax
    pad_final_kernel<<<(DD * DD + 255) / 256, 256, 0, stream>>>(W_final, b_final, Wpad, bpad);
    zero_f4_kernel<<<zeroBlocks, 256, 0, stream>>>((float4*)agg, nd4);
    gin_scatter_kernel<<<scatterBlocks, 256, 0, stream>>>(h_cur, ei, agg, E);
    gin_matmul_kernel<<<nRowTiles, 128, 0, stream>>>(
        h_cur, agg, Wpad, bpad, h_tmp, nullptr, nullptr, n, /*relu*/0, /*stats*/0);
    logsoftmax_kernel<<<lsBlocks, 256, 0, stream>>>(h_tmp, out, n);
}